// SelectiveSSM_18906446037185
// MI455X (gfx1250) — compile-verified
//
#include <hip/hip_runtime.h>

#define D_MODEL 1024
#define D_STATE 16
#define DT_RANK 64
#define SEQ     2048
#define NB      2
#define M_TOT   (NB * SEQ)              /* 4096 rows */
#define PROJ_W  (DT_RANK + 2 * D_STATE) /* 96 */
#define KCHUNK  32
#define APAD    36   /* A row stride in LDS floats: 144B = 9x16B (keeps b128 granules aligned) */
#define BPAD    34   /* Bt row stride: spreads column reads across banks, keeps b64 alignment */

typedef float v2f __attribute__((ext_vector_type(2)));
typedef float v8f __attribute__((ext_vector_type(8)));

// ---------------- WMMA f32 16x16x4 fragment helpers (CDNA5 layouts) ----------
// A (16x4, MxK): lanes 0-15 -> rows M=0..15, VGPR0/1 = K=0,1 ; lanes 16-31 -> K=2,3
__device__ __forceinline__ v2f ld_a(const float* __restrict__ A, int lda,
                                    int m0, int k0, int lane) {
  const float* p = A + (size_t)(m0 + (lane & 15)) * lda + k0 + ((lane >> 4) << 1);
  v2f a; a.x = p[0]; a.y = p[1];
  return a;
}
// B (4x16, KxN): row K striped across lanes; lanes 0-15 K=0,1 ; lanes 16-31 K=2,3
__device__ __forceinline__ v2f ld_b(const float* __restrict__ B, int ldb,
                                    int k0, int n0, int lane) {
  int col = n0 + (lane & 15);
  int kk  = k0 + ((lane >> 4) << 1);
  v2f b; b.x = B[(size_t)kk * ldb + col]; b.y = B[(size_t)(kk + 1) * ldb + col];
  return b;
}
// C/D (16x16 f32): VGPR v -> row (v + 8*(lane>=16)), col = lane&15
__device__ __forceinline__ void st_c(float* __restrict__ C, int ldc,
                                     int m0, int n0, int lane, v8f c) {
  int col = n0 + (lane & 15);
  int r0  = m0 + ((lane >> 4) << 3);
#pragma unroll
  for (int v = 0; v < 8; ++v) C[(size_t)(r0 + v) * ldc + col] = c[v];
}

#define WMMA_F32(a, b, c) \
  __builtin_amdgcn_wmma_f32_16x16x4_f32(false, (a), false, (b), (short)0, (c), false, false)

// ---------------- gfx1250 async global->LDS (ASYNCcnt) ----------------------
__device__ __forceinline__ void async_ld16(unsigned lds_off, const float* gptr) {
  unsigned long long ga = (unsigned long long)(uintptr_t)gptr;
  asm volatile("global_load_async_to_lds_b128 %0, %1, off"
               :: "v"(lds_off), "v"(ga) : "memory");
}
__device__ __forceinline__ void async_ld4(unsigned lds_off, const float* gptr) {
  unsigned long long ga = (unsigned long long)(uintptr_t)gptr;
  asm volatile("global_load_async_to_lds_b32 %0, %1, off"
               :: "v"(lds_off), "v"(ga) : "memory");
}
__device__ __forceinline__ void wait_async0() {
  asm volatile("s_wait_asynccnt 0x0" ::: "memory");
}

// ---------------- Kernel 0: A = -exp(A_log) ---------------------------------
__global__ void aneg_kernel(const float* __restrict__ Alog, float* __restrict__ An) {
  int i = blockIdx.x * blockDim.x + threadIdx.x;
  if (i < D_MODEL * D_STATE) An[i] = -__expf(Alog[i]);
}

// ---------------- Kernel 1: x_in = x @ w_in (4096x1024x1024) ----------------
// 8-wave block, 64x128 output tile, K chunked at 32, double-buffered async
// global->LDS staging. A staged row-major via b128 granules; B staged
// *transposed* (Bt[n][k]) via per-element b32 async scatter so every WMMA
// B-fragment is one aligned ds_load_b64 (no VGPR marshaling moves).
__device__ __forceinline__ void prefetch_chunk(const float* __restrict__ X,
                                               const float* __restrict__ W,
                                               int m0, int n0, int kc,
                                               unsigned asb, unsigned bsb, int tid) {
  // A tile: 64 rows x 32 k = 512 x 16B granules; 2 per lane
#pragma unroll
  for (int i = 0; i < 2; ++i) {
    int g = tid + i * 256;
    int row = g >> 3, kq = (g & 7) << 2;
    async_ld16(asb + (unsigned)((row * APAD + kq) << 2),
               X + (size_t)(m0 + row) * D_MODEL + kc + kq);
  }
  // B tile: 32 k x 128 n, transposed into LDS; global side coalesced over n
#pragma unroll
  for (int i = 0; i < 16; ++i) {
    int e = tid + i * 256;          // [0, 4096)
    int k = e >> 7, n = e & 127;
    async_ld4(bsb + (unsigned)((n * BPAD + k) << 2),
              W + (size_t)(kc + k) * D_MODEL + n0 + n);
  }
}

__global__ void __launch_bounds__(256) gemm_xin(const float* __restrict__ X,
                                                const float* __restrict__ W,
                                                float* __restrict__ O) {
  __shared__ float As[2][64 * APAD];
  __shared__ float Bs[2][128 * BPAD];
  int tid  = threadIdx.x;
  int lane = tid & 31;
  int w    = tid >> 5;
  int wm   = (w & 3) << 4;   // wave's M offset within tile: 0/16/32/48
  int wn   = (w >> 2) << 6;  // wave's N offset within tile: 0/64
  int m0   = blockIdx.y << 6;
  int n0   = blockIdx.x << 7;

  unsigned asb0 = (unsigned)(uintptr_t)&As[0][0];
  unsigned asb1 = (unsigned)(uintptr_t)&As[1][0];
  unsigned bsb0 = (unsigned)(uintptr_t)&Bs[0][0];
  unsigned bsb1 = (unsigned)(uintptr_t)&Bs[1][0];

  v8f a0 = {}, a1 = {}, a2 = {}, a3 = {};

  prefetch_chunk(X, W, m0, n0, 0, asb0, bsb0, tid);
  wait_async0();
  __syncthreads();

  const int NCH = D_MODEL / KCHUNK;  // 32
  int arow  = wm + (lane & 15);
  int koffb = (lane >> 4) << 1;
  int colb  = wn + (lane & 15);

  for (int i = 0; i < NCH; ++i) {
    int cur = i & 1;
    if (i + 1 < NCH)
      prefetch_chunk(X, W, m0, n0, (i + 1) * KCHUNK,
                     cur ? asb0 : asb1, cur ? bsb0 : bsb1, tid);
    const float* Ab = &As[cur][0];
    const float* Bb = &Bs[cur][0];
#pragma unroll
    for (int kk = 0; kk < KCHUNK; kk += 4) {
      int ka = kk + koffb;
      v2f af = *(const v2f*)(Ab + arow * APAD + ka);
      v2f b0 = *(const v2f*)(Bb + (colb     ) * BPAD + ka);
      v2f b1 = *(const v2f*)(Bb + (colb + 16) * BPAD + ka);
      v2f b2 = *(const v2f*)(Bb + (colb + 32) * BPAD + ka);
      v2f b3 = *(const v2f*)(Bb + (colb + 48) * BPAD + ka);
      a0 = WMMA_F32(af, b0, a0);
      a1 = WMMA_F32(af, b1, a1);
      a2 = WMMA_F32(af, b2, a2);
      a3 = WMMA_F32(af, b3, a3);
    }
    if (i + 1 < NCH) wait_async0();   // next chunk's LDS data has landed
    __syncthreads();                  // everyone done reading 'cur' buffer
  }

  st_c(O, D_MODEL, m0 + wm, n0 + wn,      lane, a0);
  st_c(O, D_MODEL, m0 + wm, n0 + wn + 16, lane, a1);
  st_c(O, D_MODEL, m0 + wm, n0 + wn + 32, lane, a2);
  st_c(O, D_MODEL, m0 + wm, n0 + wn + 48, lane, a3);
}

// ---------------- Kernel 2: proj = x_in @ w_x (N=96) ------------------------
__global__ void __launch_bounds__(32) gemm_proj(const float* __restrict__ Xin,
                                                const float* __restrict__ Wx,
                                                float* __restrict__ P) {
  int lane = threadIdx.x;
  int m0 = blockIdx.y << 4;
  int n0 = blockIdx.x << 4;
  v8f acc = {};
  for (int k0 = 0; k0 < D_MODEL; k0 += 4) {
    v2f af = ld_a(Xin, D_MODEL, m0, k0, lane);
    v2f bf = ld_b(Wx, PROJ_W, k0, n0, lane);
    acc = WMMA_F32(af, bf, acc);
  }
  st_c(P, PROJ_W, m0, n0, lane, acc);
}

// ---------------- Kernel 3: dt = softplus(proj[:, :64] @ w_dt + b_dt)*.099+.001
__global__ void __launch_bounds__(32) gemm_dt(const float* __restrict__ P,
                                              const float* __restrict__ Wdt,
                                              const float* __restrict__ bdt,
                                              float* __restrict__ DT) {
  int lane = threadIdx.x;
  int m0 = blockIdx.y << 4;
  int n0 = blockIdx.x << 4;
  v8f acc = {};
  for (int k0 = 0; k0 < DT_RANK; k0 += 4) {
    v2f af = ld_a(P, PROJ_W, m0, k0, lane);     // dt_low = proj cols [0,64)
    v2f bf = ld_b(Wdt, D_MODEL, k0, n0, lane);
    acc = WMMA_F32(af, bf, acc);
  }
  int col = n0 + (lane & 15);
  int r0  = m0 + ((lane >> 4) << 3);
  float bb = bdt[col];
#pragma unroll
  for (int v = 0; v < 8; ++v) {
    float val = acc[v] + bb;
    float sp  = (val > 20.f) ? val : __logf(1.f + __expf(val));  // softplus
    DT[(size_t)(r0 + v) * D_MODEL + col] = sp * 0.099f + 0.001f;
  }
}

// ---------------- Kernel 4: inp_term = ((dt*x_in) @ (-expA)) * Bgate --------
// inp_term[m,n] = Bm[m,n] * sum_d dt[m,d]*x_in[m,d]*Aneg[d,n]   (N=16)
__global__ void __launch_bounds__(32) gemm_inp(const float* __restrict__ DT,
                                               const float* __restrict__ Xin,
                                               const float* __restrict__ An,
                                               const float* __restrict__ P,
                                               float* __restrict__ INP) {
  int lane = threadIdx.x;
  int m0 = blockIdx.y << 4;
  v8f acc = {};
  int row = m0 + (lane & 15);
  for (int k0 = 0; k0 < D_MODEL; k0 += 4) {
    int kk = k0 + ((lane >> 4) << 1);
    const float* pd = DT + (size_t)row * D_MODEL + kk;
    const float* pu = Xin + (size_t)row * D_MODEL + kk;
    v2f af; af.x = pd[0] * pu[0]; af.y = pd[1] * pu[1];   // fused g = dt * u
    v2f bf = ld_b(An, D_STATE, k0, 0, lane);
    acc = WMMA_F32(af, bf, acc);
  }
  int col = lane & 15;
  int r0  = m0 + ((lane >> 4) << 3);
#pragma unroll
  for (int v = 0; v < 8; ++v) {
    int r = r0 + v;
    float bg = P[(size_t)r * PROJ_W + DT_RANK + col];     // B gate, proj cols [64,80)
    INP[(size_t)r * D_STATE + col] = acc[v] * bg;
  }
}

// ---------------- Kernel 5: sequential scan ---------------------------------
// thread = (b, d, n); h_t = exp(dt*A)*h + inp; y = sum_n h*C (+ Dp*u)
// 16-lane n-group reduction stays inside the wave32 -> no barriers.
__global__ void __launch_bounds__(256) scan_kernel(const float* __restrict__ Xin,
                                                   const float* __restrict__ DT,
                                                   const float* __restrict__ INP,
                                                   const float* __restrict__ P,
                                                   const float* __restrict__ An,
                                                   const float* __restrict__ Dp,
                                                   float* __restrict__ out) {
  int t = blockIdx.x * 256 + threadIdx.x;       // [0, 32768)
  int n = t & 15;
  int d = (t >> 4) & (D_MODEL - 1);
  int b = t >> 14;
  float a_dn = An[d * D_STATE + n];             // -exp(A_log[d,n]), per-thread const
  float dpd  = Dp[d];
  float h = 0.f;
  for (int s = 0; s < SEQ; ++s) {
    size_t base = (size_t)b * SEQ + s;
    float dtv = DT[base * D_MODEL + d];         // uniform across the 16-lane group
    float e   = __expf(dtv * a_dn);
    h = fmaf(e, h, INP[base * D_STATE + n]);
    float part = h * P[base * PROJ_W + DT_RANK + D_STATE + n];  // C gate cols [80,96)
    part += __shfl_xor(part, 1, 16);
    part += __shfl_xor(part, 2, 16);
    part += __shfl_xor(part, 4, 16);
    part += __shfl_xor(part, 8, 16);
    if (n == 0) {
      float uv = Xin[base * D_MODEL + d];
      out[base * D_MODEL + d] = part + dpd * uv;
    }
  }
}

// ---------------- Host launch ------------------------------------------------
extern "C" void kernel_launch(void* const* d_in, const int* in_sizes, int n_in,
                              void* d_out, int out_size, void* d_ws, size_t ws_size,
                              hipStream_t stream) {
  const float* x     = (const float*)d_in[0];
  const float* w_in  = (const float*)d_in[1];
  const float* w_x   = (const float*)d_in[2];
  const float* w_dt  = (const float*)d_in[3];
  const float* b_dt  = (const float*)d_in[4];
  const float* A_log = (const float*)d_in[5];
  const float* Dp    = (const float*)d_in[6];
  float* out = (float*)d_out;

  float* ws   = (float*)d_ws;
  float* xin  = ws;                                  // 4096*1024
  float* proj = xin  + (size_t)M_TOT * D_MODEL;      // 4096*96
  float* dtb  = proj + (size_t)M_TOT * PROJ_W;       // 4096*1024
  float* aneg = dtb  + (size_t)M_TOT * D_MODEL;      // 1024*16
  float* inp  = aneg + (size_t)D_MODEL * D_STATE;    // 4096*16
  (void)in_sizes; (void)n_in; (void)out_size; (void)ws_size;

  aneg_kernel<<<(D_MODEL * D_STATE + 255) / 256, 256, 0, stream>>>(A_log, aneg);
  gemm_xin <<<dim3(D_MODEL / 128, M_TOT / 64), 256, 0, stream>>>(x, w_in, xin);
  gemm_proj<<<dim3(PROJ_W / 16,  M_TOT / 16), 32, 0, stream>>>(xin, w_x, proj);
  gemm_dt  <<<dim3(D_MODEL / 16, M_TOT / 16), 32, 0, stream>>>(proj, w_dt, b_dt, dtb);
  gemm_inp <<<dim3(1,            M_TOT / 16), 32, 0, stream>>>(dtb, xin, aneg, proj, inp);
  scan_kernel<<<(NB * D_MODEL * D_STATE) / 256, 256, 0, stream>>>(xin, dtb, inp, proj,
                                                                  aneg, Dp, out);
}